// MPMModel_8907762172076
// MI455X (gfx1250) — compile-verified
//
#include <hip/hip_runtime.h>
#include <hip/hip_bf16.h>

// MPM per-particle affine update for MI455X (gfx1250).
// Memory-bound streaming kernel: 56 B/particle -> 112 MB @ 23.3 TB/s ~ 4.8 us
// (working set fits the 192 MB L2, so timed replays run at L2 bandwidth).
// CDNA5 async global->LDS copies (ASYNCcnt) triple-buffer the C/F streams per
// wave; mat/Jp are register-pipelined one tile ahead so no load latency is
// exposed between the async wait and the SVD compute.

#define WAVES_PER_BLOCK 8
#define BLOCK_THREADS   (WAVES_PER_BLOCK * 32)
#define NSTAGE          3

// ---- CDNA5 async-copy helpers (inline asm; CDNA5 ISA) --------

__device__ __forceinline__ void async_copy_b128(void* lds_dst, const void* g_src) {
    // GV mode: global_load_async_to_lds_b128 vdst(lds byte addr), vaddr64, off
    unsigned lds_off = (unsigned)(unsigned long long)lds_dst;   // low 32 bits = LDS offset
    unsigned long long ga = (unsigned long long)g_src;
    asm volatile("global_load_async_to_lds_b128 %0, %1, off"
                 :: "v"(lds_off), "v"(ga)
                 : "memory");
}

__device__ __forceinline__ void wait_async_le4() {
    asm volatile("s_wait_asynccnt 0x4" ::: "memory");
}
__device__ __forceinline__ void wait_async_le2() {
    asm volatile("s_wait_asynccnt 0x2" ::: "memory");
}
__device__ __forceinline__ void wait_async_le0() {
    asm volatile("s_wait_asynccnt 0x0" ::: "memory");
}

// ---- constants (folded from reference) ------------------------------------
#define K_DT       1e-4f
#define K_MU0      2083.3333333f      // E/(2(1+nu))
#define K_LAM0     1388.8888889f      // E*nu/((1+nu)(1-2nu))
#define K_PMASS    1.52587890625e-05f // (dx/2)^2
#define K_SCALE    (-1e-4f)           // -dt * p_vol * 4 * inv_dx^2 == -dt

__global__ __launch_bounds__(BLOCK_THREADS)
void mpm_affine_kernel(const float4* __restrict__ Cm,
                       const float4* __restrict__ Fm,
                       const int*    __restrict__ mat,
                       const float*  __restrict__ Jp,
                       float4*       __restrict__ out,
                       int n, int tiles, int wstride)
{
    __shared__ float4 sC[NSTAGE][WAVES_PER_BLOCK][32];
    __shared__ float4 sF[NSTAGE][WAVES_PER_BLOCK][32];

    const int lane = threadIdx.x & 31;
    const int wib  = threadIdx.x >> 5;
    const int gw   = blockIdx.x * WAVES_PER_BLOCK + wib;

    // Issue async copies of tile t's C and F rows into LDS stage b.
    auto issue = [&](int t, int b) {
        int pi = t * 32 + lane;
        if (pi >= n) pi = n - 1;                  // clamp partial tile (no OOB)
        async_copy_b128(&sC[b][wib][lane], Cm + pi);
        async_copy_b128(&sF[b][wib][lane], Fm + pi);
    };

    int t = gw;
    int   m_cur  = 0;
    float jp_cur = 1.0f;
    if (t < tiles) {
        issue(t, 0);                                        // stage 0 in flight
        if (t + wstride < tiles) issue(t + wstride, 1);     // stage 1 in flight
        int li = t * 32 + lane; if (li >= n) li = n - 1;
        m_cur  = mat[li];                                   // current tile scalars
        jp_cur = Jp[li];
    }
    int buf = 0;

    for (; t < tiles; t += wstride) {
        const int  t1    = t + wstride;
        const int  t2    = t + 2 * wstride;
        const bool have1 = (t1 < tiles);
        const bool have2 = (t2 < tiles);

        int   m_nxt  = 0;
        float jp_nxt = 1.0f;
        if (have2) {
            int b2 = buf + 2; if (b2 >= NSTAGE) b2 -= NSTAGE;
            issue(t2, b2);                      // keep 2 tiles in flight
        }
        if (have1) {
            int li = t1 * 32 + lane; if (li >= n) li = n - 1;
            m_nxt  = mat[li];                   // issued BEFORE the async wait:
            jp_nxt = Jp[li];                    // latency hides under compute
        }
        // Retire the current tile's 2 async ops (2/1/0 future tiles pending).
        if (have2)      wait_async_le4();
        else if (have1) wait_async_le2();
        else            wait_async_le0();

        const float4 c4 = sC[buf][wib][lane];   // ds_load_b128
        const float4 f4 = sF[buf][wib][lane];
        const int m = m_cur;
        const float jp = jp_cur;

        // ---- F' = F + dt * C@F  (row-major 2x2: [a b; c d]) ----
        const float C00 = c4.x, C01 = c4.y, C10 = c4.z, C11 = c4.w;
        float a = f4.x, b = f4.y, c = f4.z, d = f4.w;
        const float na = fmaf(K_DT, fmaf(C00, a, C01 * c), a);
        const float nb = fmaf(K_DT, fmaf(C00, b, C01 * d), b);
        const float nc = fmaf(K_DT, fmaf(C10, a, C11 * c), c);
        const float nd = fmaf(K_DT, fmaf(C10, b, C11 * d), d);
        a = na; b = nb; c = nc; d = nd;

        // ---- hardening / moduli ----
        float h = expf(10.0f * (1.0f - jp));
        if (m == 1) h = 0.3f;
        const float mu  = (m == 0) ? 0.0f : K_MU0 * h;
        const float lam = K_LAM0 * h;

        // ---- closed-form 2x2 SVD via eigen(F^T F): F = U diag(s) V^T ----
        const float p  = fmaf(a, a, c * c);
        const float q  = fmaf(b, b, d * d);
        const float r  = fmaf(a, b, c * d);
        const float ht = 0.5f * (p + q);
        const float hd = 0.5f * (p - q);
        const float disc = sqrtf(fmaf(hd, hd, r * r));
        float s1 = sqrtf(ht + disc);
        float s2 = sqrtf(fmaxf(ht - disc, 0.0f));

        // V column 1 (better-conditioned eigenvector form), column 2 = perp.
        float vx, vy;
        if (hd >= 0.0f) { vx = hd + disc; vy = r; }
        else            { vx = r;         vy = disc - hd; }
        const float nv2 = fmaf(vx, vx, vy * vy);
        if (nv2 > 1e-30f) { const float inv = rsqrtf(nv2); vx *= inv; vy *= inv; }
        else              { vx = 1.0f; vy = 0.0f; }
        // V = [vx -vy; vy vx]

        // U columns: u_i = F' v_i / s_i  (guards for degenerate s)
        float u1x, u1y, u2x, u2y;
        if (s1 > 1e-15f) {
            const float is = 1.0f / s1;
            u1x = fmaf(a, vx, b * vy) * is;
            u1y = fmaf(c, vx, d * vy) * is;
        } else { u1x = 1.0f; u1y = 0.0f; }
        if (s2 > 1e-15f) {
            const float is = 1.0f / s2;
            u2x = fmaf(b, vx, -a * vy) * is;
            u2y = fmaf(d, vx, -c * vy) * is;
        } else { u2x = -u1y; u2y = u1x; }

        // ---- snow plasticity: clamp singular values ----
        const bool snow = (m == 2);
        if (snow) {
            s1 = fminf(fmaxf(s1, 0.975f), 1.0045f);
            s2 = fminf(fmaxf(s2, 0.975f), 1.0045f);
        }
        const float J = s1 * s2;

        // ---- Fn per material (reference multiplies by V, not V^T) ----
        float Fn00 = a, Fn01 = b, Fn10 = c, Fn11 = d;
        if (snow) {
            const float t1x = s1 * vx, t1y = s1 * vy;
            const float t2x = s2 * vx, t2y = s2 * vy;
            Fn00 = fmaf(u1x, t1x,  u2x * t2y);
            Fn01 = fmaf(u2x, t2x, -u1x * t1y);
            Fn10 = fmaf(u1y, t1x,  u2y * t2y);
            Fn11 = fmaf(u2y, t2x, -u1y * t1y);
        }
        if (m == 0) {
            const float g = sqrtf(J);
            Fn00 = g; Fn01 = 0.0f; Fn10 = 0.0f; Fn11 = g;
        }

        // UV = U @ V (same buggy-V convention as reference)
        const float UV00 = fmaf(u1x, vx,  u2x * vy);
        const float UV01 = fmaf(u2x, vx, -u1x * vy);
        const float UV10 = fmaf(u1y, vx,  u2y * vy);
        const float UV11 = fmaf(u2y, vx, -u1y * vy);

        // stress = 2mu (Fn-UV) Fn^T + I * lam J (J-1); affine = -dt*stress + pmass*C
        const float D00 = Fn00 - UV00, D01 = Fn01 - UV01;
        const float D10 = Fn10 - UV10, D11 = Fn11 - UV11;
        const float tmu = 2.0f * mu;
        const float lj  = lam * J * (J - 1.0f);
        const float S00 = fmaf(tmu, fmaf(D00, Fn00, D01 * Fn01), lj);
        const float S01 =      tmu * fmaf(D00, Fn10, D01 * Fn11);
        const float S10 =      tmu * fmaf(D10, Fn00, D11 * Fn01);
        const float S11 = fmaf(tmu, fmaf(D10, Fn10, D11 * Fn11), lj);

        float4 o;
        o.x = fmaf(K_SCALE, S00, K_PMASS * C00);
        o.y = fmaf(K_SCALE, S01, K_PMASS * C01);
        o.z = fmaf(K_SCALE, S10, K_PMASS * C10);
        o.w = fmaf(K_SCALE, S11, K_PMASS * C11);
        const int idx = t * 32 + lane;
        if (idx < n) out[idx] = o;               // global_store_b128

        m_cur  = m_nxt;
        jp_cur = jp_nxt;
        buf += 1; if (buf >= NSTAGE) buf = 0;
    }
}

extern "C" void kernel_launch(void* const* d_in, const int* in_sizes, int n_in,
                              void* d_out, int out_size, void* d_ws, size_t ws_size,
                              hipStream_t stream) {
    // setup_inputs order: x(0), v(1), C(2), F(3), material(4), Jp(5).
    // x and v are unused by the reference math -> never touched (saves 32 MB).
    const float4* C   = (const float4*)d_in[2];
    const float4* F   = (const float4*)d_in[3];
    const int*    mat = (const int*)d_in[4];
    const float*  Jp  = (const float*)d_in[5];
    float4*       out = (float4*)d_out;

    const int n     = in_sizes[4];            // particle count (2,000,000)
    const int tiles = (n + 31) / 32;          // 32-particle wave tiles
    int blocks = (tiles + WAVES_PER_BLOCK - 1) / WAVES_PER_BLOCK;
    if (blocks > 2048) blocks = 2048;         // ~16K waves; ~4 tiles/wave, 2 in flight
    const int wstride = blocks * WAVES_PER_BLOCK;

    mpm_affine_kernel<<<blocks, BLOCK_THREADS, 0, stream>>>(
        C, F, mat, Jp, out, n, tiles, wstride);
}